// DAGNNConv_27659589386300
// MI455X (gfx1250) — compile-verified
//
#include <hip/hip_runtime.h>
#include <hip/hip_bf16.h>
#include <cstdint>

#define F_DIM 64
#define K_STEPS 10

typedef __attribute__((ext_vector_type(2))) float v2f;
typedef __attribute__((ext_vector_type(8))) float v8f;

// ---------------- degree / normalization ----------------

__global__ void k_init_deg(unsigned* deg, int N) {
    int i = blockIdx.x * blockDim.x + threadIdx.x;
    if (i < N) deg[i] = 1u;  // self-loop pre-counted
}

__global__ void k_deg_accum(const long long* __restrict__ col, unsigned* deg, int E) {
    int e = blockIdx.x * blockDim.x + threadIdx.x;
    if (e < E) atomicAdd(&deg[(int)col[e]], 1u);
}

__global__ void k_dinv(const unsigned* __restrict__ deg, float* __restrict__ dinv, int N) {
    int i = blockIdx.x * blockDim.x + threadIdx.x;
    if (i < N) dinv[i] = rsqrtf((float)deg[i]);   // deg >= 1 always
}

__global__ void k_edge_w(const long long* __restrict__ row, const long long* __restrict__ col,
                         const float* __restrict__ dinv, float* __restrict__ w, int E) {
    int e = blockIdx.x * blockDim.x + threadIdx.x;
    if (e < E) w[e] = dinv[(int)row[e]] * dinv[(int)col[e]];
}

__global__ void k_zero(float* __restrict__ p, size_t n) {
    size_t i = (size_t)blockIdx.x * blockDim.x + threadIdx.x;
    size_t stride = (size_t)gridDim.x * blockDim.x;
    for (; i < n; i += stride) p[i] = 0.0f;
}

// ---------------- propagation: h_next[c] += w * h_prev[r] ----------------
// blockDim = 256 : 4 edge-slots x 64 features (coalesced 64-float rows,
// per-feature GLOBAL_ATOMIC_ADD_F32 scatter; gathers are L2-resident).

__global__ void k_scatter(const float* __restrict__ hprev, float* __restrict__ hnext,
                          const long long* __restrict__ row, const long long* __restrict__ col,
                          const float* __restrict__ w, const float* __restrict__ dinv,
                          int E, int N) {
    int t    = threadIdx.x & (F_DIM - 1);
    int slot = threadIdx.x >> 6;
    long long e = (long long)blockIdx.x * 4 + slot;
    long long total = (long long)E + N;
    if (e >= total) return;
    int r, c; float wt;
    if (e < E) {
        r = (int)row[e]; c = (int)col[e]; wt = w[e];
    } else {
        int n = (int)(e - E);
        r = n; c = n;
        float di = dinv[n];
        wt = di * di;             // self-loop weight
    }
    float v = wt * hprev[(size_t)r * F_DIM + t];
    atomicAdd(&hnext[(size_t)c * F_DIM + t], v);
}

// ---------------- gate + accumulate: out += sigmoid(h.W + b) * h ----------------
// One wave32 handles 16 nodes. The 16x64 f32 tile (4KB contiguous) is staged
// into LDS with GLOBAL_LOAD_ASYNC_TO_LDS_B128 (ASYNCcnt-tracked), then both
// the WMMA A-operand feed and the gated axpy read from LDS. The 16 dot
// products h[n]·W are computed with 16 chained V_WMMA_F32_16X16X4_F32:
// A = 16x4 feature chunk, B = 4x16 with every column equal to the W chunk,
// so D[m, n] == dot(h[node m], W) for all n. fp32 end-to-end.

__global__ void k_gate_accum(const float* __restrict__ h, const float* __restrict__ Wv,
                             const float* __restrict__ bv, float* __restrict__ out, int N) {
    __shared__ float tile[8][16 * F_DIM];    // 8 waves/block * 4KB tile
    __shared__ float s_all[8 * 16];          // 8 waves/block * 16 gate pre-acts
    int lane  = threadIdx.x & 31;
    int wslot = threadIdx.x >> 5;
    int wave  = blockIdx.x * (blockDim.x >> 5) + wslot;
    int base  = wave * 16;                   // first node of this wave (N % 16 == 0)
    float* tl  = &tile[wslot][0];
    float* s_l = &s_all[wslot * 16];

    if (base < N) {
        // ---- async stage 16 rows x 64 feats = 4KB into LDS ----
        const float* src = h + (size_t)base * F_DIM;
        unsigned lds_base = (unsigned)(uintptr_t)tl;   // LDS addr = flat[31:0]
        #pragma unroll
        for (int i = 0; i < 8; ++i) {
            unsigned     loff = lds_base + (unsigned)((i * 32 + lane) * 16);
            const float* g    = src + (i * 32 + lane) * 4;
            asm volatile("global_load_async_to_lds_b128 %0, %1, off"
                         :: "v"(loff), "v"(g) : "memory");
        }
        asm volatile("s_wait_asynccnt 0" ::: "memory");

        // ---- WMMA gate dots ----
        int m  = lane & 15;                  // A-matrix row (node within tile)
        int hi = lane >> 4;                  // 0: K={0,1}, 1: K={2,3}
        const float* hrow = tl + (size_t)m * F_DIM;

        v8f d = {};
        #pragma unroll
        for (int c = 0; c < F_DIM / 4; ++c) {
            v2f a, b;
            int k0 = 4 * c + 2 * hi;
            a.x = hrow[k0 + 0];
            a.y = hrow[k0 + 1];
            b.x = Wv[k0 + 0];                // every B column is W
            b.y = Wv[k0 + 1];
            d = __builtin_amdgcn_wmma_f32_16x16x4_f32(
                    false, a, false, b, (short)0, d, false, false);
        }
        // D layout: lanes 0..15 vgpr i -> M=i, lanes 16..31 vgpr i -> M=8+i.
        // All D columns identical, so lane 0 / lane 16 hold nodes 0..7 / 8..15.
        if (lane == 0) {
            #pragma unroll
            for (int i = 0; i < 8; ++i) s_l[i] = d[i];
        }
        if (lane == 16) {
            #pragma unroll
            for (int i = 0; i < 8; ++i) s_l[8 + i] = d[i];
        }
    }
    __syncthreads();

    if (base < N) {
        float bias = bv[0];
        int nl   = lane >> 1;                // node within tile
        int half = (lane & 1) * 32;          // feature half
        float z  = s_l[nl] + bias;
        float s  = 1.0f / (1.0f + __expf(-z));
        const float* hrow = tl  + (size_t)nl * F_DIM + half;   // LDS copy
        float*       orow = out + (size_t)(base + nl) * F_DIM + half;
        #pragma unroll
        for (int i = 0; i < 32; i += 4) {
            float4 hv = *(const float4*)(hrow + i);
            float4 ov = *(float4*)(orow + i);
            ov.x += s * hv.x; ov.y += s * hv.y;
            ov.z += s * hv.z; ov.w += s * hv.w;
            *(float4*)(orow + i) = ov;
        }
    }
}

// ---------------- launch ----------------

extern "C" void kernel_launch(void* const* d_in, const int* in_sizes, int n_in,
                              void* d_out, int out_size, void* d_ws, size_t ws_size,
                              hipStream_t stream) {
    const float*      x   = (const float*)d_in[0];
    const long long*  ei  = (const long long*)d_in[1];   // int64 [2, E]
    const float*      Wv  = (const float*)d_in[2];       // [64,1]
    const float*      bv  = (const float*)d_in[3];       // [1]

    const int N = in_sizes[0] / F_DIM;
    const int E = in_sizes[1] / 2;
    const long long* row = ei;
    const long long* col = ei + E;

    const size_t NF = (size_t)N * F_DIM;

    // workspace layout
    float*    hA    = (float*)d_ws;
    float*    hB    = hA + NF;
    float*    dinv  = hB + NF;
    float*    wEdge = dinv + N;
    unsigned* deg   = (unsigned*)(wEdge + E);

    float* outp = (float*)d_out;

    const int T = 256;
    auto cdiv = [](long long a, long long b) { return (int)((a + b - 1) / b); };

    // normalization
    k_init_deg <<<cdiv(N, T), T, 0, stream>>>(deg, N);
    k_deg_accum<<<cdiv(E, T), T, 0, stream>>>(col, deg, E);
    k_dinv     <<<cdiv(N, T), T, 0, stream>>>(deg, dinv, N);
    k_edge_w   <<<cdiv(E, T), T, 0, stream>>>(row, col, dinv, wEdge, E);

    // out = 0, then accumulate gated term for k = 0 (h_0 = x)
    k_zero<<<1024, T, 0, stream>>>(outp, NF);

    const int gate_waves  = (N + 15) / 16;
    const int gate_blocks = cdiv((long long)gate_waves * 32, T);
    k_gate_accum<<<gate_blocks, T, 0, stream>>>(x, Wv, bv, outp, N);

    // K propagation steps, ping-pong between hA and hB
    const float* src = x;
    float* dstbufs[2] = {hA, hB};
    const int scat_blocks = cdiv((long long)E + N, 4);
    for (int k = 0; k < K_STEPS; ++k) {
        float* dst = dstbufs[k & 1];
        k_zero   <<<1024, T, 0, stream>>>(dst, NF);
        k_scatter<<<scat_blocks, T, 0, stream>>>(src, dst, row, col, wEdge, dinv, E, N);
        k_gate_accum<<<gate_blocks, T, 0, stream>>>(dst, Wv, bv, outp, N);
        src = dst;
    }
}